// MoE_50242527428614
// MI455X (gfx1250) — compile-verified
//
#include <hip/hip_runtime.h>
#include <math.h>

#define E 8
#define TOPK 2
#define D 768
#define F 3072
#define T_TOK 8192        // 2*64*64 tokens

#define MT 32             // token rows per block (2 TDM gathers, 2x B reuse)

typedef float v2f __attribute__((ext_vector_type(2)));
typedef float v8f __attribute__((ext_vector_type(8)));
typedef unsigned int v4u __attribute__((ext_vector_type(4)));
typedef int v8i __attribute__((ext_vector_type(8)));
typedef int v4i __attribute__((ext_vector_type(4)));

// TDM pads 4 DWORDs after every 256 DWORDs stored -> row stride 780 floats.
// Column c of row m lands at m*780 + c + 4*(c>>8). Stride 780 % 64 banks = 12
// -> the 16x4-float A-fragment reads claim all 64 LDS banks exactly once.
#define XS_STRIDE 780
#define FC 128            // F-chunk staged in LDS between FC1 and FC2
#define HS_STRIDE (FC + 4)

#define SMEM_FLOATS (MT * XS_STRIDE + MT * HS_STRIDE)
#define SMEM_BYTES  (SMEM_FLOATS * 4)

__device__ __forceinline__ int xs_idx(int m, int c) {
    return m * XS_STRIDE + c + ((c >> 8) << 2);
}

// Branch-free erf, Abramowitz-Stegun 7.1.26, |err| <= 1.5e-7 (fp32-GEMM noise).
// Hardware v_exp_f32 + v_rcp_f32 (1 ulp, fine inside a 1.5e-7 approximation).
__device__ __forceinline__ float erf_fast(float v) {
    const float a = fabsf(v);
    const float t = __builtin_amdgcn_rcpf(fmaf(0.3275911f, a, 1.0f));
    float p = fmaf(1.061405429f, t, -1.453152027f);
    p = fmaf(p, t, 1.421413741f);
    p = fmaf(p, t, -0.284496736f);
    p = fmaf(p, t, 0.254829592f);
    p = p * t;
    const float r = fmaf(-p, __expf(-a * a), 1.0f);
    return copysignf(r, v);
}

// ---------------------------------------------------------------------------
// Kernel 0: zero output (atomics accumulate from 0) and per-expert counters
// ---------------------------------------------------------------------------
__global__ void moe_init_kernel(float* __restrict__ out, int* __restrict__ counts, int n_out) {
    int i = blockIdx.x * blockDim.x + threadIdx.x;
    if (i < E) counts[i] = 0;
    for (int j = i; j < n_out; j += gridDim.x * blockDim.x) out[j] = 0.0f;
}

// ---------------------------------------------------------------------------
// Kernel 1: gating. One wave (32 lanes) per token.
// ---------------------------------------------------------------------------
__global__ __launch_bounds__(256)
void moe_gate_kernel(const float* __restrict__ x, const float* __restrict__ gw,
                     int* __restrict__ counts, int* __restrict__ tlist,
                     float* __restrict__ wlist) {
    const int wave = (blockIdx.x * blockDim.x + threadIdx.x) >> 5;
    const int lane = threadIdx.x & 31;
    if (wave >= T_TOK) return;

    const float* xr = x + (size_t)wave * D;
    float xv[D / 32];
#pragma unroll
    for (int j = 0; j < D / 32; ++j) xv[j] = xr[lane + 32 * j];

    float logit[E];
#pragma unroll
    for (int e = 0; e < E; ++e) {
        const float* g = gw + e * D;
        float s = 0.0f;
#pragma unroll
        for (int j = 0; j < D / 32; ++j) s += xv[j] * g[lane + 32 * j];
#pragma unroll
        for (int off = 16; off; off >>= 1) s += __shfl_xor(s, off, 32);
        logit[e] = s;   // replicated across lanes
    }

    // top-1 then top-2, strict '>' ascending scan == lowest-index tie break
    int e1 = 0; float l1 = logit[0];
#pragma unroll
    for (int e = 1; e < E; ++e) if (logit[e] > l1) { l1 = logit[e]; e1 = e; }
    int e2 = (e1 == 0) ? 1 : 0; float l2 = logit[e2];
#pragma unroll
    for (int e = 0; e < E; ++e) if (e != e1 && logit[e] > l2) { l2 = logit[e]; e2 = e; }

    // softmax denominator cancels: w1 = 1/(1+exp(l2-l1)), stable since l2<=l1
    const float wa = 1.0f / (1.0f + expf(l2 - l1));
    const float wb = 1.0f - wa;

    if (lane == 0) {
        int p1 = atomicAdd(&counts[e1], 1);
        tlist[e1 * T_TOK + p1] = wave; wlist[e1 * T_TOK + p1] = wa;
        int p2 = atomicAdd(&counts[e2], 1);
        tlist[e2 * T_TOK + p2] = wave; wlist[e2 * T_TOK + p2] = wb;
    }
}

// ---------------------------------------------------------------------------
// Kernel 2: fused expert MLP, WMMA f32 16x16x4 + TDM gather staging.
// grid = (E+1, ceil(T/MT)); blockIdx.x == E -> shared expert (weight 1).
// Block = 8 waves; 32-token tile (two 16-row halves sharing every B frag).
// Dynamic LDS (~114 KB, within the WGP's 320 KB).
// ---------------------------------------------------------------------------
__global__ __launch_bounds__(256)
void moe_expert_kernel(const float* __restrict__ x,
                       const float* __restrict__ w1, const float* __restrict__ b1,
                       const float* __restrict__ w2, const float* __restrict__ b2,
                       const float* __restrict__ ws1, const float* __restrict__ bs1,
                       const float* __restrict__ ws2, const float* __restrict__ bs2,
                       const int* __restrict__ counts, const int* __restrict__ tlist,
                       const float* __restrict__ wlist,
                       float* __restrict__ out) {
    extern __shared__ float smem[];
    float* Xs = smem;                       // MT x 780 (TDM-padded rows)
    float* Hs = smem + MT * XS_STRIDE;      // MT x 132
    __shared__ int   s_tid[MT];
    __shared__ float s_w[MT];

    const int  eidx       = blockIdx.x;          // 0..E-1 routed, E == shared expert
    const bool shared_exp = (eidx == E);
    const int  row0       = blockIdx.y * MT;

    const float *W1, *B1, *W2, *B2;
    int n_rows;
    if (shared_exp) {
        W1 = ws1; B1 = bs1; W2 = ws2; B2 = bs2; n_rows = T_TOK;
    } else {
        W1 = w1 + (size_t)eidx * F * D; B1 = b1 + eidx * F;
        W2 = w2 + (size_t)eidx * D * F; B2 = b2 + eidx * D;
        n_rows = counts[eidx];
    }
    if (row0 >= n_rows) return;   // uniform across block

    const int tidx = threadIdx.x;
    const int lane = tidx & 31;
    const int wid  = tidx >> 5;    // 0..7
    const int half = lane >> 4;    // 0|1 : lane group (K split in A/B frags)
    const int ln   = lane & 15;

    if (tidx < MT) {
        const int r = row0 + tidx;
        int tok; float wgt;
        if (r < n_rows) {
            if (shared_exp) { tok = r;                      wgt = 1.0f; }
            else            { tok = tlist[eidx * T_TOK + r]; wgt = wlist[eidx * T_TOK + r]; }
        } else { tok = 0; wgt = 0.0f; }   // padded rows: valid address, zero weight
        s_tid[tidx] = tok; s_w[tidx] = wgt;
    }
    __syncthreads();

    // ---- Stage gathered X tile with the Tensor Data Mover (gather mode) ----
    // Two tensor_load_to_lds ops gather 16 rows each (16-bit indices), 768 fp32
    // per row, with 4-DWORD padding every 256 DWORDs (row stride 780).
    if (wid == 0) {
        const unsigned long long ga = (unsigned long long)(uintptr_t)x;

        v8i g1;
        g1[0] = (2 << 16)          // data_size = 4B
              | (1 << 20)          // pad_enable
              | (7 << 22)          // pad_interval: 256 DWORDs
              | (3 << 25);         // pad_amount: 4 DWORDs
        g1[1] = (int)((D & 0xFFFFu) << 16);                 // tensor_dim0[15:0]
        g1[2] = (int)((D >> 16) | ((T_TOK & 0xFFFF) << 16));// tensor_dim0 hi | tensor_dim1 lo
        g1[3] = (int)((T_TOK >> 16) | (D << 16));           // tensor_dim1 hi | tile_dim0 = 768
        g1[4] = 16;                                         // tile_dim1 = 16 indices
        g1[5] = D;                                          // tensor_dim0_stride = 768
        g1[6] = 0;
        g1[7] = 0;

        const v8i gz = {0, 0, 0, 0, 0, 0, 0, 0};            // unused extra group

#pragma unroll
        for (int hblk = 0; hblk < 2; ++hblk) {              // rows 0-15, 16-31
            v4u g0;
            g0.x = 1u | (1u << 31);                         // count=1, gather_mode, 16b idx
            g0.y = (unsigned)(uintptr_t)(&Xs[hblk * 16 * XS_STRIDE]);
            g0.z = (unsigned)(ga & 0xFFFFFFFFu);
            g0.w = (unsigned)((ga >> 32) & 0x1FFFFFFu) | (2u << 30);  // type=2

            v4i g2, g3;                                     // 16 x 16-bit row indices
#pragma unroll
            for (int q = 0; q < 4; ++q) {
                const int b = hblk * 16;
                g2[q] = __builtin_amdgcn_readfirstlane(
                    (s_tid[b + 2 * q] & 0xFFFF) | (s_tid[b + 2 * q + 1] << 16));
                g3[q] = __builtin_amdgcn_readfirstlane(
                    (s_tid[b + 8 + 2 * q] & 0xFFFF) | (s_tid[b + 9 + 2 * q] << 16));
            }
            __builtin_amdgcn_tensor_load_to_lds(g0, g1, g2, g3, gz, 0);
        }
        __builtin_amdgcn_s_wait_tensorcnt(0);
    }
    __syncthreads();   // publish Xs to all waves

    const v8f vzero = {0.f, 0.f, 0.f, 0.f, 0.f, 0.f, 0.f, 0.f};
    v8f acc[6][2];     // [d-tile][row half] : 96 persistent VGPRs
#pragma unroll
    for (int j = 0; j < 6; ++j) { acc[j][0] = vzero; acc[j][1] = vzero; }

    for (int fcb = 0; fcb < F; fcb += FC) {
        // ---- FC1: 32 tokens x 16 f-columns for this wave, K = D ----
        const int f0 = fcb + wid * 16;
        v8f hacc0 = vzero, hacc1 = vzero;
        // B frag (4x16): v[j] = W1[(f0+ln)*D + k + 2*half + j], shared by both halves
        const float* w1p = W1 + (size_t)(f0 + ln) * D + 2 * half;
#pragma unroll 8
        for (int k = 0; k < D; k += 4) {
            const v2f bfr = *(const v2f*)(w1p + k);
            const v2f a0 = *(const v2f*)(&Xs[xs_idx(ln, k) + 2 * half]);
            const v2f a1 = *(const v2f*)(&Xs[xs_idx(16 + ln, k) + 2 * half]);
            hacc0 = __builtin_amdgcn_wmma_f32_16x16x4_f32(
                false, a0, false, bfr, (short)0, hacc0, false, false);
            hacc1 = __builtin_amdgcn_wmma_f32_16x16x4_f32(
                false, a1, false, bfr, (short)0, hacc1, false, false);
        }
        // bias + GELU, write both row-halves of the chunk to LDS
        const float bias1 = B1[f0 + ln];
#pragma unroll
        for (int v = 0; v < 8; ++v) {
            const int m = v + 8 * half;            // C/D layout: vgpr v, lane half
            const float h0 = hacc0[v] + bias1;
            const float h1 = hacc1[v] + bias1;
            Hs[m * HS_STRIDE + wid * 16 + ln] =
                0.5f * h0 * (1.0f + erf_fast(h0 * 0.7071067811865476f));
            Hs[(16 + m) * HS_STRIDE + wid * 16 + ln] =
                0.5f * h1 * (1.0f + erf_fast(h1 * 0.7071067811865476f));
        }
        __syncthreads();   // Hs ready for all waves

        // ---- FC2: accumulate y over this f chunk; 6 d tiles x 2 halves ----
#pragma unroll 2
        for (int kk = 0; kk < FC; kk += 4) {
            const v2f a0 = *(const v2f*)(&Hs[ln * HS_STRIDE + kk + 2 * half]);
            const v2f a1 = *(const v2f*)(&Hs[(16 + ln) * HS_STRIDE + kk + 2 * half]);
#pragma unroll
            for (int j = 0; j < 6; ++j) {
                const int d0 = (wid * 6 + j) * 16;
                const v2f bfr = *(const v2f*)(W2 + (size_t)(d0 + ln) * F + fcb + kk + 2 * half);
                acc[j][0] = __builtin_amdgcn_wmma_f32_16x16x4_f32(
                    false, a0, false, bfr, (short)0, acc[j][0], false, false);
                acc[j][1] = __builtin_amdgcn_wmma_f32_16x16x4_f32(
                    false, a1, false, bfr, (short)0, acc[j][1], false, false);
            }
        }
        __syncthreads();   // done reading Hs before next chunk overwrites it
    }

    // ---- Epilogue: out[token][d] += weight * (acc + b2[d]) ----
#pragma unroll
    for (int j = 0; j < 6; ++j) {
        const int d = (wid * 6 + j) * 16 + ln;
        const float bias2 = B2[d];
#pragma unroll
        for (int h2 = 0; h2 < 2; ++h2) {
            const v8f av = acc[j][h2];
#pragma unroll
            for (int v = 0; v < 8; ++v) {
                const int m = 16 * h2 + 8 * half + v;
                const float wgt = s_w[m];
                if (row0 + m < n_rows && wgt != 0.0f) {
                    atomicAdd(&out[(size_t)s_tid[m] * D + d], wgt * (av[v] + bias2));
                }
            }
        }
    }
}

// ---------------------------------------------------------------------------
extern "C" void kernel_launch(void* const* d_in, const int* in_sizes, int n_in,
                              void* d_out, int out_size, void* d_ws, size_t ws_size,
                              hipStream_t stream) {
    const float* x      = (const float*)d_in[0];
    const float* gate_w = (const float*)d_in[1];
    const float* w1     = (const float*)d_in[2];
    const float* b1     = (const float*)d_in[3];
    const float* w2     = (const float*)d_in[4];
    const float* b2     = (const float*)d_in[5];
    const float* ws1    = (const float*)d_in[6];
    const float* bs1    = (const float*)d_in[7];
    const float* ws2    = (const float*)d_in[8];
    const float* bs2    = (const float*)d_in[9];
    float* out = (float*)d_out;

    // workspace layout: counts | tlist | wlist   (~512 KB)
    char* ws      = (char*)d_ws;
    int*   cnts   = (int*)ws;
    int*   tlist  = (int*)(ws + 256);
    float* wlist  = (float*)(ws + 256 + sizeof(int) * E * T_TOK);

    // allow >64KB dynamic LDS (WGP supports 320KB); host-side, capture-safe
    static bool attr_set = false;
    if (!attr_set) {
        hipFuncSetAttribute((const void*)moe_expert_kernel,
                            hipFuncAttributeMaxDynamicSharedMemorySize, SMEM_BYTES);
        attr_set = true;
    }

    moe_init_kernel<<<512, 256, 0, stream>>>(out, cnts, T_TOK * D);
    moe_gate_kernel<<<T_TOK / 8, 256, 0, stream>>>(x, gate_w, cnts, tlist, wlist);

    dim3 grid(E + 1, T_TOK / MT);
    moe_expert_kernel<<<grid, 256, SMEM_BYTES, stream>>>(x, w1, b1, w2, b2,
                                                         ws1, bs1, ws2, bs2,
                                                         cnts, tlist, wlist, out);
}